// SSMLayer_36146444763587
// MI455X (gfx1250) — compile-verified
//
#include <hip/hip_runtime.h>

// SSM layer for MI455X (gfx1250, wave32).
//   step = exp(log_step); eA = (step/2)*A; G = I + eA
//   BL = inv(I - eA) = sum (eA)^k  (Horner, order 10; ||eA|| ~ 0.015)
//   Ab = BL @ G ; Bb = step * BL @ B
//   W[i,:] = C @ Ab^i        (i<128, row doubling with Ab^(2^t))
//   X[:,j] = (Ab^128)^j @ Bb (j<128, col doubling with P^(2^t))
//   K[128j+i] = (W @ X)[i,j] ; y = causal_conv(u, K)
//
// GEMMs run on the fp32 WMMA pipe. Panels are staged in LDS by the Tensor
// Data Mover with DOUBLE BUFFERING (DMA for panel i+1 overlaps WMMA on
// panel i; in-order TENSORcnt lets wave0 wait "<=2" to pin the current
// panel while the next stays in flight). Fragments read via ds_load.

typedef float v2f __attribute__((ext_vector_type(2)));
typedef float v8f __attribute__((ext_vector_type(8)));
typedef unsigned int v4u __attribute__((ext_vector_type(4)));
typedef int v4i __attribute__((ext_vector_type(4)));
typedef int v8i __attribute__((ext_vector_type(8)));

#define SSM_N 1024
#define SSM_L 16384

#define BM 64
#define BN 128
#define KC 32

// LDS offset of a __shared__ object: generic LDS addresses carry the
// aperture in the high 32 bits, LDS byte offset in the low 32.
__device__ inline unsigned lds_off(const void* p)
{
    return (unsigned)(unsigned long long)p;
}

// Issue one 2D-tile TDM load global->LDS (CDNA5 ISA ch.8 D# layout).
// tile: tile0 x tile1 elements (row-major, contiguous rows in LDS).
// tensor_dim = remaining valid extent from tile start; OOB reads => 0,
// which gives free zero-padding for the skinny doubling GEMMs.
__device__ inline void tdm_load_2d(unsigned lds_addr, const float* gptr,
                                   unsigned tile0, unsigned tile1,
                                   unsigned tdim0, unsigned tdim1,
                                   unsigned stride0)
{
    unsigned long long ga = (unsigned long long)(const void*)gptr;
    v4u g0;
    g0.x = 1u;                                   // count=1, user descriptor
    g0.y = lds_addr;                             // D#.lds_addr
    g0.z = (unsigned)(ga & 0xFFFFFFFFu);         // global_addr[31:0]
    g0.w = (unsigned)((ga >> 32) & 0x01FFFFFFu)  // global_addr[56:32]
           | (2u << 30);                         // type = 2 ("image")
    v8i g1;
    g1[0] = 0x00020000;                          // data_size=2 (4 bytes)
    g1[1] = (int)((tdim0 & 0xFFFFu) << 16);                     // tensor_dim0 lo
    g1[2] = (int)((tdim0 >> 16) | ((tdim1 & 0xFFFFu) << 16));   // dim0 hi|dim1 lo
    g1[3] = (int)((tdim1 >> 16) | (tile0 << 16));               // dim1 hi|tile0
    g1[4] = (int)(tile1 & 0xFFFFu);                             // tile1 (tile2=0)
    g1[5] = (int)stride0;                                       // dim0 stride lo
    g1[6] = 0;                                                  // stride hi, dim1 stride
    g1[7] = 0;
    v4i z4 = {0, 0, 0, 0};
#if __clang_major__ >= 23
    v8i z8 = {0, 0, 0, 0, 0, 0, 0, 0};
    __builtin_amdgcn_tensor_load_to_lds(g0, g1, z4, z4, z8, 0);
#else
    __builtin_amdgcn_tensor_load_to_lds(g0, g1, z4, z4, 0);
#endif
}

// ---------------------------------------------------------------------------
// fp32 WMMA GEMM: D(M x Nn) = A(M x K) @ B(K x Nn)  [+ I] [optional D^T store]
// Block 128 thr = 4 waves (2x2), block tile 64x128, Kc=32, wave tile 32x64.
// Double-buffered TDM panels in LDS (48 KB). Fragments per ISA 7.12.2:
//   A 16x4 : lane m = lane&15, K pair 2*(lane>>4)+{0,1}
//   B 4x16 : lane n = lane&15, same K pair split
//   C/D    : vgpr i -> row (lane>>4)*8 + i, col = lane&15
// ---------------------------------------------------------------------------
__global__ __launch_bounds__(128)
void ssm_wmma_gemm_f32(const float* __restrict__ A, int lda,
                       const float* __restrict__ B, int ldb,
                       float* __restrict__ D, int ldd,
                       int M, int Nn, int K, int addI, int storeT)
{
    __shared__ float sA[2][BM * KC];   // 2 x  8 KB: 64 x 32 row-major
    __shared__ float sB[2][KC * BN];   // 2 x 16 KB: 32 x 128 row-major

    const int tid  = threadIdx.x;
    const int wave = tid >> 5;
    const int lane = tid & 31;
    const int half = lane >> 4;
    const int lm   = lane & 15;
    const int mw   = wave >> 1;           // 0..1 : 32-row strip
    const int nw   = wave & 1;            // 0..1 : 64-col strip

    const int r0 = blockIdx.y * BM;
    const int c0 = blockIdx.x * BN;

    v8f acc[2][4];
#pragma unroll
    for (int mt = 0; mt < 2; ++mt)
#pragma unroll
        for (int nt = 0; nt < 4; ++nt)
#pragma unroll
            for (int i = 0; i < 8; ++i) acc[mt][nt][i] = 0.0f;

    // Prologue: wave0 issues panel 0 (TDM ignores EXEC; waves 1..3 skip).
    if (tid == 0) {
        tdm_load_2d(lds_off(sA[0]), A + (size_t)r0 * lda,
                    KC, BM, (unsigned)K, (unsigned)(M - r0), (unsigned)lda);
        tdm_load_2d(lds_off(sB[0]), B + c0,
                    BN, KC, (unsigned)(Nn - c0), (unsigned)K, (unsigned)ldb);
    }

    int cur = 0;
    for (int k0 = 0; k0 < K; k0 += KC, cur ^= 1) {
        if (tid == 0) {
            int kn = k0 + KC;
            if (kn < K) {   // prefetch next panel into the alternate buffer
                tdm_load_2d(lds_off(sA[cur ^ 1]), A + (size_t)r0 * lda + kn,
                            KC, BM, (unsigned)(K - kn), (unsigned)(M - r0),
                            (unsigned)lda);
                tdm_load_2d(lds_off(sB[cur ^ 1]), B + (size_t)kn * ldb + c0,
                            BN, KC, (unsigned)(Nn - c0), (unsigned)(K - kn),
                            (unsigned)ldb);
                // in-order completion: <=2 outstanding => current panel done
                __builtin_amdgcn_s_wait_tensorcnt(2);
            } else {
                __builtin_amdgcn_s_wait_tensorcnt(0);
            }
        }
        __syncthreads();

        const float* pA = sA[cur];
        const float* pB = sB[cur];
#pragma unroll
        for (int kk = 0; kk < KC; kk += 4) {
            v2f a[2];
#pragma unroll
            for (int mt = 0; mt < 2; ++mt) {
                int rl = mw * 32 + mt * 16 + lm;
                a[mt] = *(const v2f*)&pA[rl * KC + kk + 2 * half];  // ds b64
            }
#pragma unroll
            for (int nt = 0; nt < 4; ++nt) {
                int cl = nw * 64 + nt * 16 + lm;
                v2f b;
                b.x = pB[(kk + 2 * half + 0) * BN + cl];
                b.y = pB[(kk + 2 * half + 1) * BN + cl];
#pragma unroll
                for (int mt = 0; mt < 2; ++mt)
                    acc[mt][nt] = __builtin_amdgcn_wmma_f32_16x16x4_f32(
                        false, a[mt], false, b, (short)0, acc[mt][nt],
                        false, false);
            }
        }
        __syncthreads();   // all reads of `cur` done before TDM reuses it
    }

#pragma unroll
    for (int mt = 0; mt < 2; ++mt)
#pragma unroll
        for (int nt = 0; nt < 4; ++nt) {
            int cd = c0 + nw * 64 + nt * 16 + lm;
#pragma unroll
            for (int i = 0; i < 8; ++i) {
                int rd = r0 + mw * 32 + mt * 16 + half * 8 + i;
                float v = acc[mt][nt][i];
                if (addI && rd == cd) v += 1.0f;
                if (rd < M && cd < Nn) {
                    if (storeT) D[(size_t)cd * ldd + rd] = v;  // K[j*128+i]
                    else        D[(size_t)rd * ldd + cd] = v;
                }
            }
        }
}

// eA = (step/2)*A ; G = I + eA
__global__ void ssm_init_eA_G(const float* __restrict__ A,
                              const float* __restrict__ log_step,
                              float* __restrict__ eA, float* __restrict__ G,
                              int n)
{
    int idx = blockIdx.x * blockDim.x + threadIdx.x;
    if (idx >= n * n) return;
    float step = expf(log_step[0]);
    float e = 0.5f * step * A[idx];
    eA[idx] = e;
    int r = idx / n, c = idx - r * n;
    G[idx] = e + ((r == c) ? 1.0f : 0.0f);
}

// out = step * M @ v
__global__ void ssm_matvec_scaled(const float* __restrict__ Mat,
                                  const float* __restrict__ v,
                                  const float* __restrict__ log_step,
                                  float* __restrict__ out, int n)
{
    int i = blockIdx.x * blockDim.x + threadIdx.x;
    if (i >= n) return;
    float step = expf(log_step[0]);
    const float* row = Mat + (size_t)i * n;
    float s = 0.0f;
    for (int j = 0; j < n; ++j) s += row[j] * v[j];
    out[i] = step * s;
}

__global__ void ssm_copy_vec(const float* __restrict__ src,
                             float* __restrict__ dst, int n)
{
    int i = blockIdx.x * blockDim.x + threadIdx.x;
    if (i < n) dst[i] = src[i];
}

__global__ void ssm_scatter_col(const float* __restrict__ src,
                                float* __restrict__ dst, int ld, int n)
{
    int i = blockIdx.x * blockDim.x + threadIdx.x;
    if (i < n) dst[(size_t)i * ld] = src[i];
}

// y[t] = sum_{l<=t} K[l] * u[t-l], tiled through LDS. 0.13 G-MAC total.
__global__ __launch_bounds__(256)
void ssm_causal_conv(const float* __restrict__ Kf,
                     const float* __restrict__ u,
                     float* __restrict__ y, int L)
{
    __shared__ float sK[256];
    __shared__ float sU[512];
    const int t0  = blockIdx.x * 256;
    const int tid = threadIdx.x;
    const int t   = t0 + tid;
    float acc = 0.0f;

    for (int lb = 0; lb <= t0; lb += 256) {
        sK[tid] = Kf[lb + tid];
        int base = t0 - lb - 255;
        for (int k = tid; k < 511; k += 256) {
            int ui = base + k;
            sU[k] = (ui >= 0 && ui < L) ? u[ui] : 0.0f;
        }
        __syncthreads();
        if (lb + 255 <= t) {
#pragma unroll 8
            for (int dl = 0; dl < 256; ++dl)
                acc += sK[dl] * sU[tid - dl + 255];
        } else {
            int dmax = t - lb;
            if (dmax > 255) dmax = 255;
            for (int dl = 0; dl <= dmax; ++dl)
                acc += sK[dl] * sU[tid - dl + 255];
        }
        __syncthreads();
    }
    y[t] = acc;
}

extern "C" void kernel_launch(void* const* d_in, const int* in_sizes, int n_in,
                              void* d_out, int out_size, void* d_ws, size_t ws_size,
                              hipStream_t stream)
{
    (void)in_sizes; (void)n_in; (void)out_size; (void)ws_size;
    const int N = SSM_N, L = SSM_L;

    const float* u  = (const float*)d_in[0];   // (L,)
    const float* A  = (const float*)d_in[1];   // (N,N)
    const float* Bv = (const float*)d_in[2];   // (N,1)
    const float* Cv = (const float*)d_in[3];   // (1,N)
    const float* ls = (const float*)d_in[4];   // (1,1)
    float* y = (float*)d_out;                  // (L,)

    const size_t NN = (size_t)N * N;
    float* f  = (float*)d_ws;
    float* eA = f;                       // N*N
    float* G  = f + NN;                  // N*N
    float* P0 = f + 2 * NN;              // N*N
    float* P1 = f + 3 * NN;              // N*N
    float* W  = f + 4 * NN;              // 128 x N  (ld = N)
    float* X  = W + (size_t)128 * N;     // N x 128  (ld = 128)
    float* Kf = X + (size_t)128 * N;     // L
    float* Bb = Kf + L;                  // N
    // total ~17.1 MB of workspace

    auto gemm = [&](const float* Am, int lda, const float* Bm, int ldb,
                    float* Dm, int ldd, int M, int Nn, int K,
                    int addI, int storeT) {
        dim3 grid((Nn + BN - 1) / BN, (M + BM - 1) / BM);
        ssm_wmma_gemm_f32<<<grid, 128, 0, stream>>>(Am, lda, Bm, ldb, Dm, ldd,
                                                    M, Nn, K, addI, storeT);
    };

    // --- discretization ---
    ssm_init_eA_G<<<(N * N + 255) / 256, 256, 0, stream>>>(A, ls, eA, G, N);

    // BL = inv(I - eA) via Horner Neumann series: T <- I + eA@T (9 GEMMs)
    gemm(eA, N, G, N, P0, N, N, N, N, /*addI=*/1, 0);
    float* cur = P0; float* alt = P1;
    for (int it = 1; it < 9; ++it) {
        gemm(eA, N, cur, N, alt, N, N, N, N, 1, 0);
        float* tmp = cur; cur = alt; alt = tmp;
    }
    float* BL = cur;

    ssm_matvec_scaled<<<(N + 255) / 256, 256, 0, stream>>>(BL, Bv, ls, Bb, N);

    // Ab = BL @ G  -> reuse eA buffer as running power Q
    float* Q = eA;
    gemm(BL, N, G, N, Q, N, N, N, N, 0, 0);
    float* spare = (BL == P0) ? P1 : P0;

    // --- W / X by log-doubling interleaved with the squaring chain ---
    ssm_copy_vec<<<4, 256, 0, stream>>>(Cv, W, N);            // W[0,:] = C
    ssm_scatter_col<<<4, 256, 0, stream>>>(Bb, X, 128, N);    // X[:,0] = Bb

    for (int t = 0; t < 7; ++t) {          // rows: W[k:2k] = W[0:k] @ Q
        int k = 1 << t;
        gemm(W, N, Q, N, W + (size_t)k * N, N, k, N, N, 0, 0);
        gemm(Q, N, Q, N, spare, N, N, N, N, 0, 0);            // Q <- Q@Q
        float* tmp = Q; Q = spare; spare = tmp;
    }
    // here Q = Ab^128
    for (int t = 0; t < 7; ++t) {          // cols: X[:,k:2k] = Q @ X[:,0:k]
        int k = 1 << t;
        gemm(Q, N, X, 128, X + k, 128, N, k, N, 0, 0);
        if (t < 6) {
            gemm(Q, N, Q, N, spare, N, N, N, N, 0, 0);
            float* tmp = Q; Q = spare; spare = tmp;
        }
    }

    // K[j*128+i] = (W @ X)[i,j]
    gemm(W, N, X, 128, Kf, 128, 128, 128, N, 0, /*storeT=*/1);

    // y = causal_conv(u, K)
    ssm_causal_conv<<<L / 256, 256, 0, stream>>>(Kf, u, y, L);
}